// FM_pro_42056319762555
// MI455X (gfx1250) — compile-verified
//
#include <hip/hip_runtime.h>
#include <math.h>
#include <stdint.h>

typedef float v2f __attribute__((ext_vector_type(2)));
typedef float v8f __attribute__((ext_vector_type(8)));
typedef unsigned int u32x4 __attribute__((ext_vector_type(4)));
typedef int i32x4 __attribute__((ext_vector_type(4)));
typedef int i32x8 __attribute__((ext_vector_type(8)));

#define NV 64          // vectors per batch
#define DIMK 128       // vector dim
#define NPAIR 2016     // 64*63/2
#define OUTB 2018      // pairs + mean + std
#define LDS_STRIDE 132 // 128 + 4 pad -> conflict-free ds_load_b64 fragments
#define NTILE 10       // upper-tri 16x16 tiles incl. diagonal
#define THREADS 320    // 10 waves, one tile each

#if __has_builtin(__builtin_amdgcn_tensor_load_to_lds) && \
    __has_builtin(__builtin_amdgcn_s_wait_tensorcnt)
#define USE_TDM 1
#else
#define USE_TDM 0
#endif

__global__ __launch_bounds__(THREADS)
void FM_gram_triu_wmma(const float* __restrict__ in, float* __restrict__ out) {
    __shared__ __align__(16) float sA[NV][LDS_STRIDE]; // TDM-padded batch tile
    __shared__ __align__(16) float sG[NPAIR];          // triu-ordered pair values
    __shared__ float rs1[NTILE];
    __shared__ float rs2[NTILE];

    const int b = blockIdx.x;
    const float* gin = in + (size_t)b * (NV * DIMK);
    float* gout = out + (size_t)b * OUTB;

#if USE_TDM
    // One Tensor-Data-Mover DMA moves the whole 64x128 fp32 tile into LDS,
    // inserting 4 pad DWORDs after every 128-DWORD row (-> sA stride 132).
    if (threadIdx.x < 32) {  // wave 0 issues; TDM ignores EXEC, done via TENSORcnt
        const uint32_t lds_off =
            (uint32_t)(uintptr_t)(__attribute__((address_space(3))) void*)&sA[0][0];
        const uint64_t ga = (uint64_t)(uintptr_t)gin;
        // D# group0: count=1 | lds_addr | global_addr[56:0] | type=2
        const u32x4 g0 = {
            1u,
            lds_off,
            (uint32_t)(ga & 0xFFFFFFFFull),
            (uint32_t)((ga >> 32) & 0x01FFFFFFull) | (2u << 30)
        };
        // D# group1: data_size=4B (2<<16), pad_enable (1<<20),
        // pad_interval=6 (128 DWORDs), pad_amount=3 (4 DWORDs)
        // tensor_dim0=128, tensor_dim1=64, tile_dim0=128, tile_dim1=64,
        // tensor_dim0_stride=128
        const i32x8 g1 = {
            (2 << 16) | (1 << 20) | (6 << 22) | (3 << 25),
            (int)(128u << 16),  // [63:48]  tensor_dim0 lo16
            (int)(64u  << 16),  // [95:80]  tensor_dim1 lo16
            (int)(128u << 16),  // [127:112] tile_dim0
            64,                 // [143:128] tile_dim1
            128,                // [191:160] tensor_dim0_stride lo32
            0, 0
        };
        const i32x4 z4 = {0, 0, 0, 0};            // groups 2/3 unused (2D tensor)
        const i32x8 z8 = {0, 0, 0, 0, 0, 0, 0, 0};
        __builtin_amdgcn_tensor_load_to_lds(g0, g1, z4, z4, z8, 0);
        __builtin_amdgcn_s_wait_tensorcnt(0);
    }
    __syncthreads();
#else
    // Fallback: cooperative b128 staging. Row start = row*528 bytes, 16B aligned.
    for (int q = threadIdx.x; q < (NV * DIMK / 4); q += THREADS) {
        const int row = q >> 5;
        const int c4  = (q & 31) << 2;
        const float4 v = ((const float4*)gin)[q];
        *(float4*)&sA[row][c4] = v;
    }
    __syncthreads();
#endif

    const int w    = threadIdx.x >> 5;   // wave id 0..9
    const int lane = threadIdx.x & 31;
    const int half = lane >> 4;          // 0: lanes 0-15, 1: lanes 16-31
    const int lrow = lane & 15;

    // Tile list (ti<=tj): (0,0..3) (1,1..3) (2,2..3) (3,3)
    const int ti = (w < 4) ? 0 : (w < 7) ? 1 : (w < 9) ? 2 : 3;
    const int tj = (w < 4) ? w : (w < 7) ? (w - 3) : (w < 9) ? (w - 5) : 3;

    // A(16x4)/B(4x16) fragments share one layout for the symmetric Gram:
    // lane holds row lrow of its 16-row block, k = 2*half + {0,1} (+4 per step)
    const float* arow = &sA[ti * 16 + lrow][half * 2];
    const float* brow = &sA[tj * 16 + lrow][half * 2];

    v8f acc = {};
#pragma unroll
    for (int k0 = 0; k0 < DIMK; k0 += 4) {
        v2f a  = *(const v2f*)(arow + k0);
        v2f bb = *(const v2f*)(brow + k0);
        acc = __builtin_amdgcn_wmma_f32_16x16x4_f32(
            /*neg_a=*/false, a, /*neg_b=*/false, bb,
            /*c_mod=*/(short)0, acc, /*reuse_a=*/false, /*reuse_b=*/false);
    }

    // C/D layout: VGPR r holds M = r + 8*half, N = lrow.
    // Scatter pair values into LDS in triu order; accumulate sum / sumsq.
    float s1 = 0.f, s2 = 0.f;
#pragma unroll
    for (int r = 0; r < 8; ++r) {
        const int i = ti * 16 + r + half * 8;
        const int j = tj * 16 + lrow;
        const float v = acc[r];
        if (i < j) {
            const int idx = ((i * (127 - i)) >> 1) + (j - i - 1);
            sG[idx] = v;
            s1 += v;
            s2 += v * v;
        }
    }

    // wave32 reduce, then cross-wave via LDS
#pragma unroll
    for (int off = 16; off > 0; off >>= 1) {
        s1 += __shfl_xor(s1, off);
        s2 += __shfl_xor(s2, off);
    }
    if (lane == 0) { rs1[w] = s1; rs2[w] = s2; }
    __syncthreads();

    // Coalesced b64 copy-out (gout byte offset = b*8072, 8-byte aligned)
    for (int q = threadIdx.x; q < (NPAIR / 2); q += THREADS) {
        ((float2*)gout)[q] = ((const float2*)sG)[q];
    }
    if (threadIdx.x == 0) {
        float S1 = 0.f, S2 = 0.f;
#pragma unroll
        for (int t = 0; t < NTILE; ++t) { S1 += rs1[t]; S2 += rs2[t]; }
        const float mean = S1 * (1.0f / NPAIR);
        const float var  = fmaxf(S2 * (1.0f / NPAIR) - mean * mean, 0.0f);
        gout[NPAIR]     = mean;
        gout[NPAIR + 1] = sqrtf(var);
    }
}

extern "C" void kernel_launch(void* const* d_in, const int* in_sizes, int n_in,
                              void* d_out, int out_size, void* d_ws, size_t ws_size,
                              hipStream_t stream) {
    const float* in = (const float*)d_in[0];
    float* out = (float*)d_out;
    const int B = in_sizes[0] / (NV * DIMK);  // 4096
    FM_gram_triu_wmma<<<dim3(B), dim3(THREADS), 0, stream>>>(in, out);
}